// SaptioTemporalNN_large_27779848471382
// MI455X (gfx1250) — compile-verified
//
#include <hip/hip_runtime.h>
#include <math.h>

// ---- problem constants (from reference) ----
#define NXC   2000
#define NZC   4
#define NRIN  2
#define NRC   3
#define TC    16
#define KDIM  (NRC * NXC * NZC)   // 24000  (z_ctx inner dim)
#define ODIM  (NXC * NZC)         // 8000   (output features / W rows)
#define FROW  (NXC * NZC)         // 8000   floats per factors[t] slice
#define NCOL  (TC * NZC)          // 64     (t,z) columns, n = t*4+z

typedef __attribute__((ext_vector_type(2))) float v2f;
typedef __attribute__((ext_vector_type(8))) float v8f;

// Workspace layouts (interleaved k-pairs so a B fragment = one b64 load):
//   zTp   : v2f[1000][64]  : zTp[y>>1][n]   = { z[y][n],  z[y+1][n]  }
//   zctxP : v2f[12000][16] : zctxP[k>>1][t] = { zc[k][t], zc[k+1][t] }

// -------------------------------------------------------------------------
// Kernel 1: gather + transposes into pair-interleaved layouts.
// i = xz*16 + t; grid 500 x 256 = 128000 = 8000*16 exactly (no divergence).
// -------------------------------------------------------------------------
__global__ __launch_bounds__(256) void k_gather_tr(
    const float* __restrict__ factors,
    const int*   __restrict__ t_idx,
    float*       __restrict__ zTp_f,     // 2000*64 floats
    float*       __restrict__ zctxP_f) { // 24000*16 floats
    int i  = blockIdx.x * 256 + threadIdx.x;
    int t  = i & 15;
    int xz = i >> 4;                       // x*4+z, 0..7999
    float v = factors[(size_t)t_idx[t] * FROW + xz];
    // identity-relation block of z_ctx^T: k = xz
    zctxP_f[((xz >> 1) * TC + t) * 2 + (xz & 1)] = v;
    // zT pair-interleaved: y = x, n = t*4+z
    int y = xz >> 2, z = xz & 3;
    int n = t * NZC + z;
    zTp_f[((y >> 1) * NCOL + n) * 2 + (y & 1)] = v;
}

// -------------------------------------------------------------------------
// Kernel 2: relation aggregation, WMMA f32 16x16x4, y-split x4 + LDS reduce.
//   Wave wu (SGPR via readfirstlane): rr = wu&1, ys = wu>>1 (slice of 500 y).
//   Constant trip count -> scalar loop control, EXEC stays all-1s.
// grid = 125, block = 256 (8 waves).
// -------------------------------------------------------------------------
__global__ __launch_bounds__(256) void k_relagg(
    const float* __restrict__ relations,
    const float* __restrict__ zTp_f,
    float*       __restrict__ zctxP_f) {

    __shared__ float sh[8 * 4 * 8 * 32];   // [wave][nt][j][lane] = 32 KB

    const int lane = threadIdx.x & 31;
    const int wu   = __builtin_amdgcn_readfirstlane(threadIdx.x >> 5); // 0..7
    const int rr   = wu & 1;               // relation -> output r = rr+1
    const int ys   = wu >> 1;              // y slice 0..3
    const int x0   = blockIdx.x * 16;
    const int m    = lane & 15;
    const int kh   = (lane >> 4) << 1;     // 0 or 2 (K half)

    const int ybeg = ys * 500;
    const float* arow = relations + (size_t)(x0 + m) * (NRIN * NXC)
                                  + (size_t)rr * NXC + ybeg;
    // B base: pair row (ybeg+kh)/2, column m
    const v2f* bbase = (const v2f*)zTp_f + ((ybeg + kh) >> 1) * NCOL + m;

    v8f c[4] = {};
#pragma unroll 2
    for (int i = 0; i < 500; i += 4) {
        v2f a = *(const v2f*)(arow + i + kh);
        const v2f* bp = bbase + (i >> 1) * NCOL;   // (i+kh)>>1 handled in bbase
#pragma unroll
        for (int nt = 0; nt < 4; ++nt) {
            v2f b = bp[nt * 16];
            c[nt] = __builtin_amdgcn_wmma_f32_16x16x4_f32(
                false, a, false, b, (short)0, c[nt], false, false);
        }
    }

    // stash partials: sh[w][nt][j][lane]
#pragma unroll
    for (int nt = 0; nt < 4; ++nt)
#pragma unroll
        for (int j = 0; j < 8; ++j)
            sh[((wu * 4 + nt) * 8 + j) * 32 + lane] = c[nt][j];
    __syncthreads();

    // reduce over 4 y-slices; 2048 outputs / 256 threads = 8 each
    const int l = threadIdx.x & 31;
    const int q = threadIdx.x >> 5;
#pragma unroll
    for (int i = 0; i < 8; ++i) {
        int combo = q * 8 + i;             // (rr2, nt, j): 64 combos
        int rr2 = combo >> 5;
        int nt  = (combo >> 3) & 3;
        int j   = combo & 7;
        float s = 0.f;
#pragma unroll
        for (int ys2 = 0; ys2 < 4; ++ys2)
            s += sh[(((ys2 * 2 + rr2) * 4 + nt) * 8 + j) * 32 + l];
        // D layout: lane l, VGPR j -> M = j + 8*(l/16), N = l%16
        int n = nt * 16 + (l & 15);        // n = t*4+z
        int t = n >> 2, z = n & 3;
        int x = x0 + j + ((l >> 4) << 3);
        int k = (rr2 + 1) * ODIM + x * NZC + z;
        zctxP_f[((k >> 1) * TC + t) * 2 + (k & 1)] = s;
    }
}

// -------------------------------------------------------------------------
// Kernel 3: z_gen = tanh(z_ctx @ W^T + b), K-split x8 + LDS reduce.
//   Block owns one 16-row W tile (HBM stream, each byte once); wave wu covers
//   k in [wu*3000, wu*3000+3000). Inner loop: 1 b64 (W) + 1 b64 (B) + 1 WMMA.
// grid = 500, block = 256 (8 waves) -> 4000 waves streaming W_dyn.
// -------------------------------------------------------------------------
#define KSLICE 3000                        // 24000 / 8 waves

__global__ __launch_bounds__(256) void k_gemm_tanh(
    const float* __restrict__ W,
    const float* __restrict__ b_dyn,
    const float* __restrict__ zctxP_f,
    float*       __restrict__ out) {

    __shared__ float sh[8 * 8 * 32];       // [wave][j][lane] = 8 KB

    const int lane = threadIdx.x & 31;
    const int wu   = __builtin_amdgcn_readfirstlane(threadIdx.x >> 5); // 0..7
    const int o0   = blockIdx.x * 16;
    const int m    = lane & 15;
    const int kh   = (lane >> 4) << 1;

    const int kbeg = wu * KSLICE;
    const float* wbase = W + (size_t)(o0 + m) * KDIM + kbeg + kh;
    const v2f*   bbase = (const v2f*)zctxP_f + ((kbeg + kh) >> 1) * TC + m;

    v8f c = {};
    for (int i = 0; i < KSLICE; i += 40) {          // 75 uniform outer iters
        __builtin_prefetch(wbase + i + 640, 0, 1);  // global_prefetch W ahead
#pragma unroll
        for (int ku = 0; ku < 40; ku += 4) {
            v2f a = *(const v2f*)(wbase + i + ku);
            v2f b = bbase[((i + ku) >> 1) * TC];
            c = __builtin_amdgcn_wmma_f32_16x16x4_f32(
                false, a, false, b, (short)0, c, false, false);
        }
    }

    // stash partials: sh[w][j][lane]
#pragma unroll
    for (int j = 0; j < 8; ++j)
        sh[(wu * 8 + j) * 32 + lane] = c[j];
    __syncthreads();

    // reduce over 8 K-slices; 256 outputs, one per thread
    const int j = threadIdx.x >> 5;
    const int l = threadIdx.x & 31;
    float s = 0.f;
#pragma unroll
    for (int w2 = 0; w2 < 8; ++w2)
        s += sh[(w2 * 8 + j) * 32 + l];
    // D: lane l, VGPR j -> M(=o offset) = j + 8*(l/16), N(=t) = l%16
    int o = o0 + j + ((l >> 4) << 3);
    int t = l & 15;
    out[(size_t)t * ODIM + o] = tanhf(s + b_dyn[o]);
}

// -------------------------------------------------------------------------
extern "C" void kernel_launch(void* const* d_in, const int* in_sizes, int n_in,
                              void* d_out, int out_size, void* d_ws, size_t ws_size,
                              hipStream_t stream) {
    const float* relations = (const float*)d_in[0];  // (2000, 2, 2000)
    const float* factors   = (const float*)d_in[1];  // (200, 2000, 4)
    const float* W_dyn     = (const float*)d_in[2];  // (8000, 24000)
    const float* b_dyn     = (const float*)d_in[3];  // (8000,)
    const int*   t_idx     = (const int*)d_in[4];    // (16,) int32
    float* out = (float*)d_out;                      // (16, 8000)

    float* zTp   = (float*)d_ws;                     // 2000*64 floats = 512 KB
    float* zctxP = zTp + NXC * NCOL;                 // 24000*16 floats = 1.536 MB

    k_gather_tr<<<(TC * ODIM) / 256, 256, 0, stream>>>(factors, t_idx, zTp, zctxP);
    k_relagg<<<NXC / 16, 256, 0, stream>>>(relations, zTp, zctxP);
    k_gemm_tanh<<<ODIM / 16, 256, 0, stream>>>(W_dyn, b_dyn, zctxP, out);
}